// T3GNN_49271864819962
// MI455X (gfx1250) — compile-verified
//
#include <hip/hip_runtime.h>
#include <math.h>

typedef float v2f __attribute__((ext_vector_type(2)));
typedef float v8f __attribute__((ext_vector_type(8)));

#define HIDDEN 128

// ---------------------------------------------------------------------------
// WMMA fp32 GEMM:  C[M,N] = act( A[M,K1]@B1[K1,N] (+ A2[M,K2]@B2[K2,N]) + bias )
// One wave per 16x64 output strip. M % 16 == 0, N % 64 == 0, K % 4 == 0.
// Fragment layouts per CDNA5 ISA 7.12.2 (16x16x4 f32):
//   A (16x4):  lanes 0-15 -> M, VGPR0/1 = K=(0,1) low half, K=(2,3) high half
//   B (4x16):  lanes(&15) -> N, VGPR0/1 = K=(0,1) low half, K=(2,3) high half
//   C/D:       VGPR r -> M = r (+8 for lanes 16-31), N = lane & 15
// ---------------------------------------------------------------------------
__global__ __launch_bounds__(32) void wmma_gemm_f32(
    const float* __restrict__ A,  int K1, const float* __restrict__ B1,
    const float* __restrict__ A2, int K2, const float* __restrict__ B2,
    const float* __restrict__ bias, float* __restrict__ C,
    int M, int N, int relu)
{
    const int lane = threadIdx.x;      // 0..31
    const int half = lane >> 4;        // 0 / 1
    const int l16  = lane & 15;
    const int tileM   = blockIdx.y;
    const int colBase = blockIdx.x * 64;
    const size_t row  = (size_t)tileM * 16 + l16;

    v8f c0 = {}, c1 = {}, c2 = {}, c3 = {};

    // ---- pass 1: A @ B1 ----
    {
        const float* Arow = A + row * (size_t)K1;
        for (int k = 0; k < K1; k += 4) {
            const int ka = k + half * 2;
            v2f a = *(const v2f*)(Arow + ka);                  // A[row,ka..ka+1]
            const float* Bp = B1 + (size_t)ka * N + colBase + l16;
            v2f b0 = { Bp[0],  Bp[(size_t)N] };
            v2f b1 = { Bp[16], Bp[(size_t)N + 16] };
            v2f b2 = { Bp[32], Bp[(size_t)N + 32] };
            v2f b3 = { Bp[48], Bp[(size_t)N + 48] };
            c0 = __builtin_amdgcn_wmma_f32_16x16x4_f32(false, a, false, b0, (short)0, c0, false, false);
            c1 = __builtin_amdgcn_wmma_f32_16x16x4_f32(false, a, false, b1, (short)0, c1, false, false);
            c2 = __builtin_amdgcn_wmma_f32_16x16x4_f32(false, a, false, b2, (short)0, c2, false, false);
            c3 = __builtin_amdgcn_wmma_f32_16x16x4_f32(false, a, false, b3, (short)0, c3, false, false);
        }
    }
    // ---- pass 2 (fused concat): A2 @ B2 ----
    if (A2 != nullptr) {
        const float* Arow = A2 + row * (size_t)K2;
        for (int k = 0; k < K2; k += 4) {
            const int ka = k + half * 2;
            v2f a = *(const v2f*)(Arow + ka);
            const float* Bp = B2 + (size_t)ka * N + colBase + l16;
            v2f b0 = { Bp[0],  Bp[(size_t)N] };
            v2f b1 = { Bp[16], Bp[(size_t)N + 16] };
            v2f b2 = { Bp[32], Bp[(size_t)N + 32] };
            v2f b3 = { Bp[48], Bp[(size_t)N + 48] };
            c0 = __builtin_amdgcn_wmma_f32_16x16x4_f32(false, a, false, b0, (short)0, c0, false, false);
            c1 = __builtin_amdgcn_wmma_f32_16x16x4_f32(false, a, false, b1, (short)0, c1, false, false);
            c2 = __builtin_amdgcn_wmma_f32_16x16x4_f32(false, a, false, b2, (short)0, c2, false, false);
            c3 = __builtin_amdgcn_wmma_f32_16x16x4_f32(false, a, false, b3, (short)0, c3, false, false);
        }
    }

    float bv0 = 0.f, bv1 = 0.f, bv2 = 0.f, bv3 = 0.f;
    if (bias != nullptr) {
        bv0 = bias[colBase + l16];
        bv1 = bias[colBase + 16 + l16];
        bv2 = bias[colBase + 32 + l16];
        bv3 = bias[colBase + 48 + l16];
    }
    #pragma unroll
    for (int r = 0; r < 8; ++r) {
        const size_t m = (size_t)tileM * 16 + half * 8 + r;
        float* Cp = C + m * N + colBase + l16;
        float v0 = c0[r] + bv0;
        float v1 = c1[r] + bv1;
        float v2 = c2[r] + bv2;
        float v3 = c3[r] + bv3;
        if (relu) {
            v0 = fmaxf(v0, 0.f); v1 = fmaxf(v1, 0.f);
            v2 = fmaxf(v2, 0.f); v3 = fmaxf(v3, 0.f);
        }
        Cp[0] = v0; Cp[16] = v1; Cp[32] = v2; Cp[48] = v3;
    }
}

// ---------------------------------------------------------------------------
// Degree / normalization kernels
// ---------------------------------------------------------------------------
__global__ void deg_init_kernel(float* __restrict__ deg, int n) {
    int i = blockIdx.x * blockDim.x + threadIdx.x;
    if (i < n) deg[i] = 1.0f;   // self-loop weight
}

__global__ void deg_accum_kernel(const int* __restrict__ dst, float* __restrict__ deg, int E) {
    int i = blockIdx.x * blockDim.x + threadIdx.x;
    if (i < E) atomicAdd(&deg[dst[i]], 1.0f);
}

__global__ void deg_rsqrt_kernel(float* __restrict__ deg, int n) {
    int i = blockIdx.x * blockDim.x + threadIdx.x;
    if (i < n) deg[i] = rsqrtf(deg[i]);   // in place -> dinv
}

// AGG[i,:] = HW[i,:] * dinv[i]^2   (self-loop term, also zero-initializes AGG)
__global__ void gcn_self_init_kernel(const float* __restrict__ HW,
                                     const float* __restrict__ dinv,
                                     float* __restrict__ AGG, int total) {
    int i = blockIdx.x * blockDim.x + threadIdx.x;
    if (i < total) {
        float d = dinv[i / HIDDEN];
        AGG[i] = HW[i] * d * d;
    }
}

// one wave per edge: AGG[dst,:] += HW[src,:] * dinv[src]*dinv[dst]
__global__ __launch_bounds__(256) void gcn_scatter_kernel(
    const float* __restrict__ HW, const int* __restrict__ src,
    const int* __restrict__ dst, const float* __restrict__ dinv,
    float* __restrict__ AGG, int E)
{
    int e    = (blockIdx.x * blockDim.x + threadIdx.x) >> 5;
    int lane = threadIdx.x & 31;
    if (e >= E) return;
    int s = src[e], d = dst[e];
    float w = dinv[s] * dinv[d];
    const float* hs = HW + (size_t)s * HIDDEN;
    float* ad = AGG + (size_t)d * HIDDEN;
    #pragma unroll
    for (int i = 0; i < HIDDEN; i += 32)
        atomicAdd(&ad[i + lane], hs[i + lane] * w);
}

// OUT = relu(AGG + bias)
__global__ void bias_relu_kernel(const float* __restrict__ AGG,
                                 const float* __restrict__ b,
                                 float* __restrict__ OUT, int total) {
    int i = blockIdx.x * blockDim.x + threadIdx.x;
    if (i < total) OUT[i] = fmaxf(AGG[i] + b[i % HIDDEN], 0.f);
}

// one wave per label edge: score = dot(e1[a]*e1[b], pw[:,0]+pw[:,1]) + pb0+pb1
__global__ __launch_bounds__(256) void link_score_kernel(
    const float* __restrict__ emb, const int* __restrict__ eli,
    const float* __restrict__ pw, const float* __restrict__ pb,
    float* __restrict__ score, int L)
{
    int l    = (blockIdx.x * blockDim.x + threadIdx.x) >> 5;
    int lane = threadIdx.x & 31;
    if (l >= L) return;
    int a = eli[l];
    int b = eli[L + l];
    const float* ea = emb + (size_t)a * HIDDEN;
    const float* eb = emb + (size_t)b * HIDDEN;
    float sum = 0.f;
    #pragma unroll
    for (int i = 0; i < HIDDEN; i += 32) {
        int d = i + lane;
        float pws = pw[2 * d] + pw[2 * d + 1];
        sum += ea[d] * eb[d] * pws;
    }
    #pragma unroll
    for (int off = 16; off > 0; off >>= 1)
        sum += __shfl_down(sum, off, 32);
    if (lane == 0) score[l] = sum + pb[0] + pb[1];
}

// ---------------------------------------------------------------------------
extern "C" void kernel_launch(void* const* d_in, const int* in_sizes, int n_in,
                              void* d_out, int out_size, void* d_ws, size_t ws_size,
                              hipStream_t stream)
{
    const float* x     = (const float*)d_in[0];
    const float* prev0 = (const float*)d_in[1];
    const float* prev1 = (const float*)d_in[2];
    const int*   eidx  = (const int*)d_in[3];
    const int*   elidx = (const int*)d_in[4];
    const float* W1  = (const float*)d_in[5];
    const float* b1  = (const float*)d_in[6];
    const float* W2  = (const float*)d_in[7];
    const float* b2  = (const float*)d_in[8];
    const float* gw0 = (const float*)d_in[9];
    const float* gb0 = (const float*)d_in[10];
    const float* gw1 = (const float*)d_in[11];
    const float* gb1 = (const float*)d_in[12];
    const float* uw0 = (const float*)d_in[13];
    const float* ub0 = (const float*)d_in[14];
    const float* uw1 = (const float*)d_in[15];
    const float* ub1 = (const float*)d_in[16];
    const float* pw  = (const float*)d_in[17];
    const float* pb  = (const float*)d_in[18];

    const int NN   = in_sizes[1] / HIDDEN;     // 50000 nodes (prev0 is [N,128])
    const int INF  = in_sizes[0] / NN;         // 512
    const int MID  = in_sizes[6];              // 256 (b1)
    const int E    = in_sizes[3] / 2;          // 600000
    const int L    = in_sizes[4] / 2;          // 200000

    const int* src = eidx;
    const int* dst = eidx + E;

    // workspace carve-up (floats)
    float* ws   = (float*)d_ws;
    float* dinv = ws;                                   // NN
    float* H1   = dinv + NN;                            // NN * MID
    float* H2   = H1 + (size_t)NN * MID;                // NN * HIDDEN (aliased as HR)
    float* HW   = H2 + (size_t)NN * HIDDEN;             // NN * HIDDEN
    float* AGG  = HW + (size_t)NN * HIDDEN;             // NN * HIDDEN
    float* HR   = H2;                                   // alias: H2 dead once HW computed

    float* score = (float*)d_out;                       // L
    float* e0    = score + L;                           // NN * HIDDEN
    float* e1    = e0 + (size_t)NN * HIDDEN;            // NN * HIDDEN

    const int totalH = NN * HIDDEN;
    const dim3 blk(256);
    const int gN  = (NN + 255) / 256;
    const int gE  = (E + 255) / 256;
    const int gH  = (totalH + 255) / 256;
    const int gEw = (E * 32 + 255) / 256;   // one wave per edge
    const int gLw = (L * 32 + 255) / 256;   // one wave per label edge

    auto gemm = [&](const float* A, int K1, const float* B1,
                    const float* A2, int K2, const float* B2,
                    const float* bias, float* C, int N, int relu) {
        dim3 grid(N / 64, NN / 16);
        wmma_gemm_f32<<<grid, dim3(32), 0, stream>>>(A, K1, B1, A2, K2, B2, bias, C, NN, N, relu);
    };

    // degree -> dinv
    deg_init_kernel<<<gN, blk, 0, stream>>>(dinv, NN);
    deg_accum_kernel<<<gE, blk, 0, stream>>>(dst, dinv, E);
    deg_rsqrt_kernel<<<gN, blk, 0, stream>>>(dinv, NN);

    // preprocess MLP
    gemm(x,  INF, W1, nullptr, 0, nullptr, b1, H1, MID,    1);  // H1 = relu(x@W1+b1)
    gemm(H1, MID, W2, nullptr, 0, nullptr, b2, H2, HIDDEN, 1);  // H2 = relu(H1@W2+b2)

    // ---- layer 0 ----
    gemm(H2, HIDDEN, gw0, nullptr, 0, nullptr, nullptr, HW, HIDDEN, 0);   // HW = H2@gw0
    gcn_self_init_kernel<<<gH, blk, 0, stream>>>(HW, dinv, AGG, totalH);
    gcn_scatter_kernel<<<gEw, blk, 0, stream>>>(HW, src, dst, dinv, AGG, E);
    bias_relu_kernel<<<gH, blk, 0, stream>>>(AGG, gb0, HR, totalH);
    // e0 = [HR ; prev0] @ uw0 + ub0   (fused dual-K GEMM)
    gemm(HR, HIDDEN, uw0, prev0, HIDDEN, uw0 + HIDDEN * HIDDEN, ub0, e0, HIDDEN, 0);

    // ---- layer 1 ----
    gemm(e0, HIDDEN, gw1, nullptr, 0, nullptr, nullptr, HW, HIDDEN, 0);   // HW = e0@gw1
    gcn_self_init_kernel<<<gH, blk, 0, stream>>>(HW, dinv, AGG, totalH);
    gcn_scatter_kernel<<<gEw, blk, 0, stream>>>(HW, src, dst, dinv, AGG, E);
    bias_relu_kernel<<<gH, blk, 0, stream>>>(AGG, gb1, HR, totalH);
    gemm(HR, HIDDEN, uw1, prev1, HIDDEN, uw1 + HIDDEN * HIDDEN, ub1, e1, HIDDEN, 0);

    // link scoring
    link_score_kernel<<<gLw, blk, 0, stream>>>(e1, elidx, pw, pb, score, L);
}